// FarthestPointSample_9732395892840
// MI455X (gfx1250) — compile-verified
//
#include <hip/hip_runtime.h>
#include <stdint.h>

// FPS: input [8, 65536, 3] f32 -> (idx [8,1024] i32, sampled [8,1024,3] f32)
#define BATCH     8
#define NPTS      65536
#define NSAMP     1024
#define NTHREADS  1024
#define WPB       4                          // cooperating workgroups per batch
#define SLICE     (NPTS / WPB)               // 16384 points per WG
#define TILE_PTS  4096
#define NT        (SLICE / TILE_PTS)         // 4 tiles per sweep per WG
#define PPT       (SLICE / NTHREADS)         // 16 min_dist entries per lane (VGPRs)
#define JPT       (TILE_PTS / NTHREADS)      // 4 points per lane per tile
#define TILE_BYTES  (TILE_PTS * 3 * 4)       // 49152 B
#define TILE_DWORDS (TILE_PTS * 3)           // 12288

static_assert(NT * JPT == PPT, "tiling mismatch");

typedef uint32_t u32;
typedef u32 v4u __attribute__((ext_vector_type(4)));
typedef u32 v8u __attribute__((ext_vector_type(8)));

// Issue one TDM DMA: 48KB contiguous tile of f32 coords, global -> LDS.
// D# built per CDNA5 ISA 8.3/8.4 (2 SGPR groups, <=2D tensor, data_size=4B).
__device__ __forceinline__ void tdm_load_tile(const float* gsrc, u32 lds_byte_off) {
  uint64_t ga  = (uint64_t)(uintptr_t)gsrc;
  u32 glo = (u32)__builtin_amdgcn_readfirstlane((int)(ga & 0xffffffffull));
  u32 ghi = (u32)__builtin_amdgcn_readfirstlane((int)(ga >> 32));
  u32 lds = (u32)__builtin_amdgcn_readfirstlane((int)lds_byte_off);

  v4u g0;
  g0[0] = 1u;                                   // count=1 (valid), user mode
  g0[1] = lds;                                  // lds_addr (bytes, WG-relative)
  g0[2] = glo;                                  // global_addr[31:0]
  g0[3] = (ghi & 0x01ffffffu) | (2u << 30);     // global_addr[56:32] | type=2

  v8u g1;
  g1[0] = 2u << 16;                                  // wg_mask=0, data_size=2 (4B)
  g1[1] = ((u32)TILE_DWORDS & 0xffffu) << 16;        // tensor_dim0[15:0]
  g1[2] = ((u32)TILE_DWORDS >> 16) | (1u << 16);     // tensor_dim0[31:16] | tensor_dim1=1
  g1[3] = ((u32)TILE_DWORDS & 0xffffu) << 16;        // tensor_dim1 hi=0 | tile_dim0=12288
  g1[4] = 1u;                                        // tile_dim1=1, tile_dim2=0
  g1[5] = (u32)TILE_DWORDS;                          // tensor_dim0_stride[31:0]
  g1[6] = ((u32)TILE_DWORDS & 0xffffu) << 16;        // stride0 hi=0 | stride1[15:0]
  g1[7] = 0u;                                        // stride1 hi=0

  asm volatile("tensor_load_to_lds %0, %1" :: "s"(g0), "s"(g1) : "memory");
}

extern "C" __global__ __launch_bounds__(NTHREADS)
void fps_tdm_kernel(const float* __restrict__ pts,
                    int* __restrict__ out_idx,
                    float* __restrict__ out_xyz,
                    unsigned long long* __restrict__ slots,   // [BATCH*NSAMP] argmax keys
                    unsigned int* __restrict__ cnts) {        // [BATCH*NSAMP] arrival counters
  extern __shared__ char smem[];
  float* redV = (float*)(smem + 2 * TILE_BYTES);
  int*   redI = (int*)(redV + 32);
  float* sQ   = (float*)(redI + 32);   // broadcast: current farthest coords
  int*   sQI  = (int*)(sQ + 4);        // broadcast: current farthest index

  const int tid = threadIdx.x;
  const int w   = blockIdx.x;          // 0..WPB-1  (slice within batch)
  const int b   = blockIdx.y;          // 0..BATCH-1
  const float* bpts = pts + (size_t)b * NPTS * 3;
  const float* spts = bpts + (size_t)w * SLICE * 3;

  // Per-lane running min squared distance, kept in VGPRs (all indices static).
  float md[PPT];
#pragma unroll
  for (int m = 0; m < PPT; ++m) md[m] = 1e10f;   // BIG from reference

  if (tid == 0) {
    sQI[0] = 0;
    sQ[0] = bpts[0]; sQ[1] = bpts[1]; sQ[2] = bpts[2];
  }
  __syncthreads();

  const u32 smem_base = (u32)(uintptr_t)smem;

  for (int i = 0; i < NSAMP; ++i) {
    const float qx = sQ[0], qy = sQ[1], qz = sQ[2];
    if (w == 0 && tid == 0) {                    // one WG per batch emits outputs
      out_idx[b * NSAMP + i] = sQI[0];
      float* o = out_xyz + ((size_t)b * NSAMP + i) * 3;
      o[0] = qx; o[1] = qy; o[2] = qz;
    }
    if (i == NSAMP - 1) break;   // last emitted index needs no further update

    // Preload tile 0 into buffer 0 (wave 0 issues the DMA; EXEC ignored by TDM).
    if (tid < 32) tdm_load_tile(spts, smem_base);

    float bv = -1.0f; int bi = 0;

#pragma unroll
    for (int t = 0; t < NT; ++t) {
      if (tid < 32) __builtin_amdgcn_s_wait_tensorcnt(0);  // buf[t&1] DMA done
      __syncthreads();                                     // publish tile to all waves
      if ((t + 1 < NT) && tid < 32)                        // prefetch next tile
        tdm_load_tile(spts + (size_t)(t + 1) * TILE_PTS * 3,
                      smem_base + (u32)((t + 1) & 1) * TILE_BYTES);

      const float* buf = (const float*)(smem + (t & 1) * TILE_BYTES);
#pragma unroll
      for (int j = 0; j < JPT; ++j) {
        const int lp = j * NTHREADS + tid;          // point within tile
        const float x = buf[lp * 3 + 0];
        const float y = buf[lp * 3 + 1];
        const float z = buf[lp * 3 + 2];
        const float dx = x - qx, dy = y - qy, dz = z - qz;
        float d = dx * dx;
        d = fmaf(dy, dy, d);
        d = fmaf(dz, dz, d);
        const int m = t * JPT + j;                  // static after unroll -> VGPR
        const float nm = fminf(md[m], d);
        md[m] = nm;
        const int g = w * SLICE + t * TILE_PTS + lp; // batch-global point index
        if (nm > bv) { bv = nm; bi = g; }           // keep earliest max
      }
    }

    // Wave32 argmax reduction (max value, earliest index on ties).
#pragma unroll
    for (int off = 16; off >= 1; off >>= 1) {
      const float ov = __shfl_xor(bv, off, 32);
      const int   oi = __shfl_xor(bi, off, 32);
      if (ov > bv || (ov == bv && oi < bi)) { bv = ov; bi = oi; }
    }
    if ((tid & 31) == 0) { redV[tid >> 5] = bv; redI[tid >> 5] = bi; }
    __syncthreads();

    if (tid < 32) {
      bv = redV[tid]; bi = redI[tid];
#pragma unroll
      for (int off = 16; off >= 1; off >>= 1) {
        const float ov = __shfl_xor(bv, off, 32);
        const int   oi = __shfl_xor(bi, off, 32);
        if (ov > bv || (ov == bv && oi < bi)) { bv = ov; bi = oi; }
      }
      if (tid == 0) {
        // Cross-WG argmax combine: distances are >= 0 so the fp32 bit pattern
        // is order-preserving as u32; inverted index -> smallest index on ties.
        unsigned long long* slot = slots + (size_t)b * NSAMP + i;
        unsigned int*       cnt  = cnts  + (size_t)b * NSAMP + i;
        const unsigned long long key =
            ((unsigned long long)__float_as_uint(bv) << 32) |
            (unsigned int)(0xFFFFFFFFu - (unsigned int)bi);
        atomicMax(slot, key);                                     // device-scope u64 max
        __hip_atomic_fetch_add(cnt, 1u, __ATOMIC_RELEASE,
                               __HIP_MEMORY_SCOPE_AGENT);         // publish + arrive
        while (__hip_atomic_load(cnt, __ATOMIC_ACQUIRE,
                                 __HIP_MEMORY_SCOPE_AGENT) < WPB)
          __builtin_amdgcn_s_sleep(1);                            // polite spin
        const unsigned long long kk =
            __hip_atomic_load(slot, __ATOMIC_RELAXED, __HIP_MEMORY_SCOPE_AGENT);
        const int win = (int)(0xFFFFFFFFu - (unsigned int)kk);
        sQI[0] = win;
        const float* p = bpts + (size_t)win * 3;    // L2-resident fetch of winner
        sQ[0] = p[0]; sQ[1] = p[1]; sQ[2] = p[2];
      }
    }
    __syncthreads();
  }
}

extern "C" void kernel_launch(void* const* d_in, const int* in_sizes, int n_in,
                              void* d_out, int out_size, void* d_ws, size_t ws_size,
                              hipStream_t stream) {
  (void)in_sizes; (void)n_in; (void)out_size; (void)ws_size;
  const float* pts = (const float*)d_in[0];
  int*   out_idx = (int*)d_out;                    // first 8*1024 int32 slots
  float* out_xyz = (float*)d_out + BATCH * NSAMP;  // then 8*1024*3 f32

  unsigned long long* slots = (unsigned long long*)d_ws;
  unsigned int* cnts = (unsigned int*)((char*)d_ws +
                       (size_t)BATCH * NSAMP * sizeof(unsigned long long));
  // Zero per-iteration argmax slots + arrival counters (capturable async memset).
  hipMemsetAsync(d_ws, 0,
                 (size_t)BATCH * NSAMP * (sizeof(unsigned long long) + sizeof(unsigned int)),
                 stream);

  const size_t shmem = 2 * TILE_BYTES + 512;       // double-buffer + reduction area
  fps_tdm_kernel<<<dim3(WPB, BATCH), dim3(NTHREADS), shmem, stream>>>(
      pts, out_idx, out_xyz, slots, cnts);
}